// NeighborRoutingAgg_53927609368716
// MI455X (gfx1250) — compile-verified
//
#include <hip/hip_runtime.h>
#include <math.h>

// ---------------------------------------------------------------------------
// NeighborRoutingAgg on MI455X (gfx1250).
// Per row n: G = Z Z^T via v_wmma_f32_16x16x4_f32 (exact fp32 GEMM,
// M=32,N=32,K=128 as 2x2 tiles x 32 K-chunks), routing iterations in the
// 32-dim space, final u = Z^T p + x in fp32.
// ---------------------------------------------------------------------------

typedef __attribute__((ext_vector_type(2))) float v2f;
typedef __attribute__((ext_vector_type(8))) float v8f;

#define DVEC      128
#define MNB       32
#define Z_STRIDE  132                 // floats; 528B row: 16B aligned, bank-friendly
#define G_STRIDE  36                  // floats; 144B row: 16B aligned
#define Z_OFF     0
#define X_OFF     (MNB * Z_STRIDE)            // 4224
#define G_OFF     (X_OFF + DVEC)              // 4352
#define P_OFF     (G_OFF + MNB * G_STRIDE)    // 5504
#define WAVE_FLOATS (P_OFF + MNB)             // 5536 floats = 22144 B per wave

// -------------------------- kernel 1: inverse norms -------------------------
__global__ __launch_bounds__(256) void rnorm_kernel(const float* __restrict__ x,
                                                    float* __restrict__ rn,
                                                    int n_total) {
    const int lane = threadIdx.x & 31;
    const int wave = threadIdx.x >> 5;
    const int row  = blockIdx.x * 8 + wave;
    if (row >= n_total) return;
    float4 v = *(const float4*)(x + (size_t)row * DVEC + lane * 4);
    float ss = v.x * v.x + v.y * v.y + v.z * v.z + v.w * v.w;
    #pragma unroll
    for (int off = 16; off >= 1; off >>= 1) ss += __shfl_xor(ss, off, 32);
    if (lane == 0) rn[row] = 1.0f / fmaxf(sqrtf(ss), 1e-12f);
}

// -------------------------- kernel 2: routing -------------------------------
__global__ __launch_bounds__(64) void routing_kernel(const float* __restrict__ x,
                                                     const int* __restrict__ nb,
                                                     const float* __restrict__ rnorm,
                                                     float* __restrict__ out,
                                                     int n_total) {
    __shared__ __align__(16) float smem[2 * WAVE_FLOATS];
    const int lane = threadIdx.x & 31;
    const int wave = threadIdx.x >> 5;
    const int n    = blockIdx.x * 2 + wave;
    if (n >= n_total) return;                 // uniform per wave

    float* S = smem + wave * WAVE_FLOATS;     // per-wave private LDS region
    float* Z = S + Z_OFF;
    float* X = S + X_OFF;
    float* G = S + G_OFF;
    float* P = S + P_OFF;

    // ---- per-lane neighbor index & its inverse norm (lane == m) ----
    const int   jm = nb[(size_t)n * MNB + lane] - 1;   // 1-based -> 0-based
    const float rj = rnorm[jm];

    // ---- stage normalized x_n ----
    {
        const float rn = rnorm[n];
        float4 v = *(const float4*)(x + (size_t)n * DVEC + lane * 4);
        float4 w = make_float4(v.x * rn, v.y * rn, v.z * rn, v.w * rn);
        *(float4*)(X + lane * 4) = w;
    }

    // ---- stage normalized Z rows (one 512B coalesced row per step) ----
    for (int m = 0; m < MNB; ++m) {
        const int   j  = __shfl(jm, m, 32);
        const float rm = __shfl(rj, m, 32);
        float4 v = *(const float4*)(x + (size_t)j * DVEC + lane * 4);
        float4 w = make_float4(v.x * rm, v.y * rm, v.z * rm, v.w * rm);
        *(float4*)(Z + m * Z_STRIDE + lane * 4) = w;
    }
    // single-wave region: LDS ops from the same wave complete in order.

    // ---- Gram: G = Z Z^T via WMMA f32 16x16x4, 2x2 tiles, 32 K-chunks ----
    // A-tile layout (16x4 f32): lane<16 -> {m=l, k=kc,kc+1}; lane>=16 -> {m=l-16, k=kc+2,kc+3}
    // B-tile of Z^T columns has the identical per-lane layout -> reuse registers.
    v8f C00 = {}, C01 = {}, C10 = {}, C11 = {};
    const int lsub = lane & 15;
    const int koff = (lane >> 4) * 2;
    #pragma unroll 4
    for (int kc = 0; kc < DVEC; kc += 4) {
        v2f t0 = *(const v2f*)(Z + lsub * Z_STRIDE + kc + koff);
        v2f t1 = *(const v2f*)(Z + (16 + lsub) * Z_STRIDE + kc + koff);
        C00 = __builtin_amdgcn_wmma_f32_16x16x4_f32(false, t0, false, t0, (short)0, C00, false, false);
        C01 = __builtin_amdgcn_wmma_f32_16x16x4_f32(false, t0, false, t1, (short)0, C01, false, false);
        C10 = __builtin_amdgcn_wmma_f32_16x16x4_f32(false, t1, false, t0, (short)0, C10, false, false);
        C11 = __builtin_amdgcn_wmma_f32_16x16x4_f32(false, t1, false, t1, (short)0, C11, false, false);
    }

    // ---- spill G tiles to LDS (C/D layout: VGPR r -> rows r / r+8) ----
    {
        const int rbase = (lane < 16) ? 0 : 8;
        const int col   = lane & 15;
        #pragma unroll
        for (int r = 0; r < 8; ++r) {
            G[(rbase + r) * G_STRIDE + col]            = C00[r];
            G[(rbase + r) * G_STRIDE + 16 + col]       = C01[r];
            G[(16 + rbase + r) * G_STRIDE + col]       = C10[r];
            G[(16 + rbase + r) * G_STRIDE + 16 + col]  = C11[r];
        }
    }

    // ---- b = Z x  (lane m owns b[m]) ----
    float bm = 0.0f;
    #pragma unroll 8
    for (int d = 0; d < DVEC; d += 4) {
        float4 zv = *(const float4*)(Z + lane * Z_STRIDE + d);
        float4 xv = *(const float4*)(X + d);
        bm += zv.x * xv.x + zv.y * xv.y + zv.z * xv.z + zv.w * xv.w;
    }

    // ---- routing iterations in 32-dim space ----
    // p0 = softmax(0) = 1/32.  For it = 0..2:
    //   a = G p + b ; ||u||^2 = p.a + p.b + 1 ; s = squash/||u|| ; p = softmax(s*a)
    float pm = 1.0f / 32.0f;
    for (int it = 0; it < 3; ++it) {
        P[lane] = pm;
        float am = bm;
        #pragma unroll 8
        for (int mp = 0; mp < MNB; ++mp)
            am += G[lane * G_STRIDE + mp] * P[mp];

        float pa = pm * am, pb = pm * bm;
        #pragma unroll
        for (int off = 16; off >= 1; off >>= 1) {
            pa += __shfl_xor(pa, off, 32);
            pb += __shfl_xor(pb, off, 32);
        }
        const float sq = pa + pb + 1.0f;
        const float s  = (sq / (sq + 1.0f)) / fmaxf(sqrtf(sq), 1e-12f);

        float t  = s * am;
        float mx = t;
        #pragma unroll
        for (int off = 16; off >= 1; off >>= 1) mx = fmaxf(mx, __shfl_xor(mx, off, 32));
        float e  = expf(t - mx);
        float se = e;
        #pragma unroll
        for (int off = 16; off >= 1; off >>= 1) se += __shfl_xor(se, off, 32);
        pm = e / se;
    }
    P[lane] = pm;

    // ---- final u = Z^T p + x  (lane owns d = 4*lane .. 4*lane+3) ----
    float4 acc = *(const float4*)(X + lane * 4);
    #pragma unroll 8
    for (int m = 0; m < MNB; ++m) {
        const float pv = P[m];
        float4 zv = *(const float4*)(Z + m * Z_STRIDE + lane * 4);
        acc.x += zv.x * pv; acc.y += zv.y * pv; acc.z += zv.z * pv; acc.w += zv.w * pv;
    }
    *(float4*)(out + (size_t)n * DVEC + lane * 4) = acc;
}

// ---------------------------------------------------------------------------
extern "C" void kernel_launch(void* const* d_in, const int* in_sizes, int n_in,
                              void* d_out, int out_size, void* d_ws, size_t ws_size,
                              hipStream_t stream) {
    (void)n_in; (void)out_size; (void)ws_size;
    const float* x  = (const float*)d_in[0];
    const int*   nb = (const int*)d_in[1];
    float*       out = (float*)d_out;
    float*       rn  = (float*)d_ws;          // n floats of scratch
    const int n = in_sizes[0] / DVEC;

    rnorm_kernel<<<(n + 7) / 8, 256, 0, stream>>>(x, rn, n);
    routing_kernel<<<(n + 1) / 2, 64, 0, stream>>>(x, nb, rn, out, n);
}